// VQVAE_33749853012114
// MI455X (gfx1250) — compile-verified
//
#include <hip/hip_runtime.h>
#include <hip/hip_bf16.h>
#include <math.h>

typedef __attribute__((ext_vector_type(2))) float v2f;
typedef __attribute__((ext_vector_type(8))) float v8f;

static __device__ __forceinline__ v8f wmma_f32_k4(v2f a, v2f b, v8f c) {
    // D = A(16x4 f32) * B(4x16 f32) + C(16x16 f32): exact fp32 matrix pipe
    return __builtin_amdgcn_wmma_f32_16x16x4_f32(false, a, false, b, (short)0, c, false, false);
}

// CDNA5 async global->LDS copy: 16B per lane, tracked by ASYNCcnt.
static __device__ __forceinline__ void async_copy_b128(unsigned lds_addr, const void* gaddr) {
    unsigned long long ga = (unsigned long long)gaddr;
    asm volatile("global_load_async_to_lds_b128 %0, %1, off"
                 :: "v"(lds_addr), "v"(ga) : "memory");
}
static __device__ __forceinline__ void async_wait0() {
    asm volatile("s_wait_asynccnt 0" ::: "memory");
}
static __device__ __forceinline__ unsigned lds_addr_of(const void* p) {
    return (unsigned)(unsigned long long)p;   // LDS aperture: low 32 bits = LDS offset
}

// ---------------------------------------------------------------------------
// Generic WMMA implicit-GEMM conv (stride S, pad P).
// Block = (batch, out-row, row-split). (COUT/16) waves; wave w owns couts
// [16w,16w+16). Input patch for the row-tile + K-chunks of weights live in LDS;
// weights staged with global_load_async_to_lds_b128 (layout [co][K] is
// contiguous per row). LDS weight rows padded to KC+4 -> conflict-free
// ds_load_b64 B-fragments; B loaded once per K-step, reused across NT tiles.
// ACT: 0 = ReLU, 1 = sigmoid. COUT may be zero-padded above COUT_REAL.
// ---------------------------------------------------------------------------
template<int CIN, int COUT, int COUT_REAL, int KH, int KW, int S, int PAD,
         int IH, int IW, int KC, int SPLIT, int ACT>
__global__ __launch_bounds__((COUT / 16) * 32) void k_conv2d_wmma(
    const float* __restrict__ in, const float* __restrict__ wgt,
    const float* __restrict__ bias, float* __restrict__ out)
{
    constexpr int K    = CIN * KH * KW;
    constexpr int OH   = (IH + 2 * PAD - KH) / S + 1;
    constexpr int OW   = (IW + 2 * PAD - KW) / S + 1;
    constexpr int TW   = OW / SPLIT;            // output cols per block
    constexpr int PW   = (TW - 1) * S + KW;     // staged patch width
    constexpr int NT   = TW / 16;               // 16-pixel tiles per block
    constexpr int KCP  = KC + 4;                // padded LDS weight row stride
    constexpr int KC4  = KC / 4;                // 16B chunks per weight row
    constexpr int NTHR = (COUT / 16) * 32;

    __shared__ float Pin[CIN * KH * PW];
    __shared__ float Wl[COUT * KCP];
    __shared__ float Bl[COUT];

    const int blk = blockIdx.x;
    const int b   = blk / (OH * SPLIT);
    const int rem = blk % (OH * SPLIT);
    const int oy  = rem / SPLIT;
    const int xb  = rem % SPLIT;
    const int x0  = xb * TW;                    // output col base
    const int tid = threadIdx.x;

    for (int i = tid; i < COUT; i += NTHR) Bl[i] = bias[i];
    // Pin[ci][r][c] = in[b][ci][oy*S-PAD+r][x0*S-PAD+c]  (zero padded)
    for (int i = tid; i < CIN * KH * PW; i += NTHR) {
        const int ci = i / (KH * PW);
        const int r2 = i % (KH * PW);
        const int r  = r2 / PW;
        const int c  = r2 % PW;
        const int iy = oy * S - PAD + r;
        const int ix = x0 * S - PAD + c;
        float v = 0.f;
        if (iy >= 0 && iy < IH && ix >= 0 && ix < IW)
            v = in[((b * CIN + ci) * IH + iy) * IW + ix];
        Pin[i] = v;
    }

    const int lane = tid & 31;
    const int wv   = tid >> 5;
    const int lo   = lane & 15;
    const int half = lane >> 4;
    const int coB  = wv * 16;
    const int wrow = (coB + lo) * KCP;

    v8f acc[NT];
#pragma unroll
    for (int t = 0; t < NT; t++)
#pragma unroll
        for (int j = 0; j < 8; j++) acc[t][j] = 0.f;

    for (int kc = 0; kc < K; kc += KC) {
        __syncthreads();
        // async-stage weight chunk: Wl[co][kl] <- wgt[co][kc+kl], 16B per lane
        for (int i = tid; i < COUT * KC4; i += NTHR) {
            const int co = i / KC4;
            const int q4 = (i % KC4) * 4;
            async_copy_b128(lds_addr_of(&Wl[co * KCP + q4]),
                            wgt + co * K + kc + q4);
            if (kc + KC < K)  // prefetch next chunk into GL2
                __builtin_prefetch(wgt + co * K + kc + KC + q4, 0, 1);
        }
        async_wait0();
        __syncthreads();

        for (int kk = 0; kk < KC; kk += 4) {
            const int k0l = kk + half * 2;       // lane's K pair (A/B striping)
            const int kg0 = kc + k0l;
            const int kg1 = kg0 + 1;
            const int ci0 = kg0 / (KH * KW), re0 = kg0 % (KH * KW);
            const int r0  = re0 / KW,        kx0 = re0 % KW;
            const int ci1 = kg1 / (KH * KW), re1 = kg1 % (KH * KW);
            const int r1  = re1 / KW,        kx1 = re1 % KW;
            const int base0 = (ci0 * KH + r0) * PW + kx0;
            const int base1 = (ci1 * KH + r1) * PW + kx1;
            v2f bb;
            bb.x = Wl[wrow + k0l];
            bb.y = Wl[wrow + k0l + 1];
#pragma unroll
            for (int t = 0; t < NT; t++) {       // reuse B across pixel tiles
                const int ms = (t * 16 + lo) * S;
                v2f a;
                a.x = Pin[base0 + ms];
                a.y = Pin[base1 + ms];
                acc[t] = wmma_f32_k4(a, bb, acc[t]);
            }
        }
    }

    const int co = coB + lo;
    if (co < COUT_REAL) {
        const float bv = Bl[co];
#pragma unroll
        for (int t = 0; t < NT; t++)
#pragma unroll
            for (int j = 0; j < 8; j++) {
                const int m = t * 16 + j + half * 8;
                float v = acc[t][j] + bv;
                if (ACT == 0) v = v > 0.f ? v : 0.f;
                else          v = 1.f / (1.f + expf(-v));
                out[((b * COUT_REAL + co) * OH + oy) * OW + x0 + m] = v;
            }
    }
}

// ---------------------------------------------------------------------------
// Generic WMMA ConvTranspose2d k4 s2 p1 + ReLU.
// out[oy][ox] = sum in[(oy+1-ky)/2][(ox+1-kx)/2] * w[ci][co][ky][kx].
// Per output-column parity p there are exactly 2 valid kx (and 2 valid ky per
// oy) -> per-parity GEMM with K = CIN*4. One block = (b, oy). Weights are a
// gather (parity-dependent taps), staged manually into [co][K]-padded LDS.
// ---------------------------------------------------------------------------
template<int CIN, int COUT, int IH, int IW, int KC>
__global__ __launch_bounds__((COUT / 16) * 32) void k_deconv2d_wmma(
    const float* __restrict__ in, const float* __restrict__ wgt,
    const float* __restrict__ bias, float* __restrict__ out)
{
    constexpr int OH  = IH * 2, OW = IW * 2;
    constexpr int K   = CIN * 4;          // per-parity GEMM depth
    constexpr int CW  = IW + 2;           // patch cols (ix in [-1, IW])
    constexpr int NT  = (OW / 2) / 16;    // 16-pixel tiles per parity
    constexpr int KCP = KC + 4;
    constexpr int NTHR = (COUT / 16) * 32;

    __shared__ float Pin[CIN * 2 * CW];
    __shared__ float Wl[COUT * KCP];
    __shared__ float Bl[COUT];

    const int b  = blockIdx.x / OH;
    const int oy = blockIdx.x % OH;
    const int q  = oy & 1;
    const int R  = oy >> 1;
    const int tid = threadIdx.x;

    for (int i = tid; i < COUT; i += NTHR) Bl[i] = bias[i];
    // Pin[ci][r][c] = in[b][ci][R+q-1+r][c-1], r in {0,1}
    for (int i = tid; i < CIN * 2 * CW; i += NTHR) {
        const int ci = i / (2 * CW);
        const int r2 = i % (2 * CW);
        const int r  = r2 / CW;
        const int c  = r2 % CW;
        const int iy = R + q - 1 + r;
        const int ix = c - 1;
        float v = 0.f;
        if (iy >= 0 && iy < IH && ix >= 0 && ix < IW)
            v = in[((b * CIN + ci) * IH + iy) * IW + ix];
        Pin[i] = v;
    }

    const int lane = tid & 31;
    const int wv   = tid >> 5;
    const int lo   = lane & 15;
    const int half = lane >> 4;
    const int coB  = wv * 16;
    const int wrow = (coB + lo) * KCP;

    for (int p = 0; p < 2; p++) {          // output column parity
        v8f acc[NT];
#pragma unroll
        for (int t = 0; t < NT; t++)
#pragma unroll
            for (int j = 0; j < 8; j++) acc[t][j] = 0.f;

        for (int kc = 0; kc < K; kc += KC) {
            __syncthreads();
            // Wl[co][kl] = wgt[ci][co][ky][kx], k = ci*4 + ky_i*2 + kx_i
            for (int i = tid; i < COUT * KC; i += NTHR) {
                const int co = i / KC;
                const int kl = i % KC;
                const int k  = kc + kl;
                const int ci = k >> 2;
                const int ky = (1 - q) + 2 * ((k >> 1) & 1);
                const int kx = (1 - p) + 2 * (k & 1);
                Wl[co * KCP + kl] = wgt[((ci * COUT + co) * 4 + ky) * 4 + kx];
            }
            __syncthreads();

            for (int kk = 0; kk < KC; kk += 4) {
                const int k0l = kk + half * 2;
                const int kg0 = kc + k0l;
                const int kg1 = kg0 + 1;
                const int ci0 = kg0 >> 2, ky0 = (kg0 >> 1) & 1, kx0 = kg0 & 1;
                const int ci1 = kg1 >> 2, ky1 = (kg1 >> 1) & 1, kx1 = kg1 & 1;
                const int base0 = (ci0 * 2 + (1 - ky0)) * CW + p + 1 - kx0;
                const int base1 = (ci1 * 2 + (1 - ky1)) * CW + p + 1 - kx1;
                v2f bb;
                bb.x = Wl[wrow + k0l];
                bb.y = Wl[wrow + k0l + 1];
#pragma unroll
                for (int t = 0; t < NT; t++) {
                    const int m = t * 16 + lo;
                    v2f a;
                    a.x = Pin[base0 + m];
                    a.y = Pin[base1 + m];
                    acc[t] = wmma_f32_k4(a, bb, acc[t]);
                }
            }
        }

        const int co = coB + lo;
        const float bv = Bl[co];
#pragma unroll
        for (int t = 0; t < NT; t++)
#pragma unroll
            for (int j = 0; j < 8; j++) {
                const int m = t * 16 + j + half * 8;
                const int ox = 2 * m + p;
                float v = acc[t][j] + bv;
                v = v > 0.f ? v : 0.f;
                out[((b * COUT + co) * OH + oy) * OW + ox] = v;
            }
    }
}

// ---------------------------------------------------------------------------
// 1x1 conv 64->8 + VQ (argmin over 512 codes, D=8) fused.
// Writes z (NCHW, for loss), zq flat F[n*8+d] (decoder input via the
// reference's NCHW-reshape bug), and float-cast indices into d_out.
// ---------------------------------------------------------------------------
__global__ __launch_bounds__(256) void k_enc3_vq(
    const float* __restrict__ h2, const float* __restrict__ w3,
    const float* __restrict__ b3, const float* __restrict__ cb,
    float* __restrict__ z_ws, float* __restrict__ zq_ws,
    float* __restrict__ idx_out)
{
    __shared__ float Wl[8 * 64];
    __shared__ float Bl[8];
    __shared__ float Cb[512 * 8];
    const int tid = threadIdx.x;
    for (int i = tid; i < 512; i += 256) Wl[i] = w3[i];
    if (tid < 8) Bl[tid] = b3[tid];
    for (int i = tid; i < 4096; i += 256) Cb[i] = cb[i];
    __syncthreads();

    const int n = blockIdx.x * 256 + tid;      // NHWC pixel id, 0..131071
    const int w = n & 63;
    const int h = (n >> 6) & 63;
    const int b = n >> 12;

    float z[8];
#pragma unroll
    for (int d = 0; d < 8; d++) z[d] = Bl[d];
    for (int ci = 0; ci < 64; ci++) {
        const float hv = h2[((b * 64 + ci) * 64 + h) * 64 + w];
#pragma unroll
        for (int d = 0; d < 8; d++) z[d] = fmaf(hv, Wl[d * 64 + ci], z[d]);
    }
#pragma unroll
    for (int d = 0; d < 8; d++)
        z_ws[((b * 8 + d) * 64 + h) * 64 + w] = z[d];

    float best = 3.4e38f;
    int bi = 0;
    for (int k = 0; k < 512; k++) {
        float dd = 0.f;
#pragma unroll
        for (int d = 0; d < 8; d++) {
            const float df = z[d] - Cb[k * 8 + d];
            dd = fmaf(df, df, dd);
        }
        if (dd < best) { best = dd; bi = k; }   // first-min tie-break (argmin)
    }
#pragma unroll
    for (int d = 0; d < 8; d++) zq_ws[n * 8 + d] = Cb[bi * 8 + d];
    idx_out[n] = (float)bi;
}

// ---------------------------------------------------------------------------
// vq_loss = 1.25 * mean((z - zq)^2) over 1,048,576 elems (NCHW pairing).
// ---------------------------------------------------------------------------
__global__ void k_zero_loss(float* p) { if (threadIdx.x == 0) p[0] = 0.f; }

__global__ __launch_bounds__(256) void k_vq_loss(
    const float* __restrict__ z, const float* __restrict__ zq,
    float* __restrict__ loss)
{
    __shared__ float red[256];
    float s = 0.f;
    for (int i = blockIdx.x * 256 + threadIdx.x; i < 1048576; i += gridDim.x * 256) {
        const float d = z[i] - zq[i];
        s = fmaf(d, d, s);
    }
    red[threadIdx.x] = s;
    __syncthreads();
    for (int st = 128; st > 0; st >>= 1) {
        if (threadIdx.x < st) red[threadIdx.x] += red[threadIdx.x + st];
        __syncthreads();
    }
    if (threadIdx.x == 0)
        atomicAdd(loss, red[0] * (1.25f / 1048576.f));
}

// ---------------------------------------------------------------------------
// Build zero-padded (3->16 couts) weights/bias for the final conv so it can
// run through the WMMA kernel. Rebuilt every call (ws is not persistent).
// ---------------------------------------------------------------------------
__global__ __launch_bounds__(256) void k_pad_dec2(
    const float* __restrict__ w, const float* __restrict__ b,
    float* __restrict__ wpad, float* __restrict__ bpad)
{
    const int i = blockIdx.x * 256 + threadIdx.x;
    if (i < 16 * 288) {
        const int co = i / 288, k = i % 288;
        wpad[i] = (co < 3) ? w[co * 288 + k] : 0.f;
    }
    if (i < 16) bpad[i] = (i < 3) ? b[i] : 0.f;
}

// ---------------------------------------------------------------------------
// Launcher: full VQ-VAE forward on `stream`.
// ---------------------------------------------------------------------------
extern "C" void kernel_launch(void* const* d_in, const int* in_sizes, int n_in,
                              void* d_out, int out_size, void* d_ws, size_t ws_size,
                              hipStream_t stream)
{
    const float* x   = (const float*)d_in[0];
    const float* ew1 = (const float*)d_in[1];
    const float* eb1 = (const float*)d_in[2];
    const float* ew2 = (const float*)d_in[3];
    const float* eb2 = (const float*)d_in[4];
    const float* ew3 = (const float*)d_in[5];
    const float* eb3 = (const float*)d_in[6];
    const float* cb  = (const float*)d_in[7];
    const float* dw1 = (const float*)d_in[8];
    const float* db1 = (const float*)d_in[9];
    const float* tw1 = (const float*)d_in[10];
    const float* tb1 = (const float*)d_in[11];
    const float* tw2 = (const float*)d_in[12];
    const float* tb2 = (const float*)d_in[13];
    const float* dw2 = (const float*)d_in[14];
    const float* db2 = (const float*)d_in[15];

    float* ws = (float*)d_ws;
    // workspace layout (floats)
    float* h1   = ws;                          // [32,32,128,128] 16,777,216
    float* h2   = ws + 16777216;               // [32,64,64,64]    8,388,608
    float* z    = ws + 25165824;               // [32,8,64,64]     1,048,576
    float* zq   = ws + 26214400;               // flat F[n*8+d]    1,048,576
    float* t1   = ws + 27262976;               // [32,64,128,128] 33,554,432
    float* t2   = ws + 60817408;               // [32,32,256,256] 67,108,864
    float* wpad = ws + 127926272;              // 16*288 padded dec2 weights
    float* bpad = wpad + 16 * 288;             // 16 padded dec2 bias
    float* d1   = h1;                          // [32,64,64,64] aliases dead h1

    float* x_recon  = (float*)d_out;           // 6,291,456
    float* loss_out = x_recon + 6291456;       // 1
    float* idx_out  = x_recon + 6291457;       // 131,072 (float-cast indices)

    // encoder (all WMMA implicit-GEMM)
    k_conv2d_wmma<3, 32, 32, 4, 4, 2, 1, 256, 256, 48, 1, 0>
        <<<32 * 128, 64, 0, stream>>>(x, ew1, eb1, h1);
    k_conv2d_wmma<32, 64, 64, 4, 4, 2, 1, 128, 128, 128, 1, 0>
        <<<32 * 64, 128, 0, stream>>>(h1, ew2, eb2, h2);
    // 1x1 conv + VQ (argmin, gather with faithful NCHW-reshape bug)
    k_enc3_vq<<<512, 256, 0, stream>>>(h2, ew3, eb3, cb, z, zq, idx_out);
    // vq loss (zero each call: graph replays must not accumulate)
    k_zero_loss<<<1, 64, 0, stream>>>(loss_out);
    k_vq_loss<<<1024, 256, 0, stream>>>(z, zq, loss_out);
    // decoder
    k_conv2d_wmma<8, 64, 64, 3, 3, 1, 1, 64, 64, 72, 1, 0>
        <<<32 * 64, 128, 0, stream>>>(zq, dw1, db1, d1);
    k_deconv2d_wmma<64, 64, 64, 64, 64>
        <<<32 * 128, 128, 0, stream>>>(d1, tw1, tb1, t1);
    k_deconv2d_wmma<64, 32, 128, 128, 64>
        <<<32 * 256, 64, 0, stream>>>(t1, tw2, tb2, t2);
    k_pad_dec2<<<18, 256, 0, stream>>>(dw2, db2, wpad, bpad);
    k_conv2d_wmma<32, 16, 3, 3, 3, 1, 1, 256, 256, 96, 2, 1>
        <<<32 * 256 * 2, 32, 0, stream>>>(t2, wpad, bpad, x_recon);
}